// NaiveModel_76862734729914
// MI455X (gfx1250) — compile-verified
//
#include <hip/hip_runtime.h>
#include <math.h>

// ===================== problem constants (match reference) =====================
constexpr int   B_   = 2;
constexpr int   C_   = 8;
constexpr int   H_   = 64;
constexpr int   NF_  = 512;              // NFFT
constexpr int   HOP_ = 256;
constexpr int   LEN_ = 48000;
constexpr int   F_   = NF_ / 2 + 1;      // 257
constexpr int   T_   = 1 + LEN_ / HOP_;  // 188
constexpr int   FT_  = F_ * T_;          // 48316
constexpr float EPS_   = 1e-4f;
constexpr float SLOPE_ = 0.01f;

typedef __attribute__((ext_vector_type(2))) float v2f;
typedef __attribute__((ext_vector_type(8))) float v8f;

static inline int cdiv(int a, int b) { return (a + b - 1) / b; }

// ===================== device math helpers =====================
__device__ __forceinline__ float d_lrelu(float x) { return x > 0.f ? x : SLOPE_ * x; }
__device__ __forceinline__ float d_snap(float x) {
  if (x >= 0.f && x <  EPS_) return  EPS_;
  if (x <  0.f && x > -EPS_) return -EPS_;
  return x;
}
__device__ __forceinline__ float d_modexp(float x) {
  const float em = 148.41315910257660342f; // exp(5)
  return (x > 5.f) ? ((x - 5.f) * em + em) : expf(fminf(x, 5.f));
}
__device__ __forceinline__ float d_sigmoid(float x) { return 1.f / (1.f + expf(-x)); }

// ===================== generic batched WMMA GEMM =====================
// C[z][m, n] = alpha * sum_k A[z][m, k] * B[z][k, n + bOff] + beta * C + bias
// A element   : A[m*lda + k*KSA]
// B element   : B[k*ldb + (n + bOff)]   (contribution zeroed unless 0 <= n+bOff < bMax)
// batch z     : zi = z % zInner, zo = z / zInner; ptr += zi*s?i + zo*s?o
//
// Strategy (MI455X): no divergent loads in the hot loop. Addresses are clamped
// (always legal); only the B operand is value-masked (needed for the shifted
// temporal-conv window). Out-of-range M rows / N cols merely produce garbage in
// output elements that are store-masked anyway. Each wave computes a 16x32 C
// tile (two N-subtiles sharing one A operand); a 4-wave block covers 32x64.
struct GP {
  const float* A; const float* B; float* C;
  int M, N, K;
  int lda, ksA, ldb, ldc;
  int zInner;
  long long sAi, sAo, sBi, sBo, sCi, sCo;
  const float* bias; const float* bias2; float b2s;
  float alpha, beta;
  int bOff, bMax;
};

template <int KSA>
__global__ __launch_bounds__(128) void k_gemm_wmma(GP p) {
  const int lane = threadIdx.x & 31;
  const int wid  = threadIdx.x >> 5;
  const int m0 = blockIdx.y * 32 + (wid >> 1) * 16;
  const int n0 = blockIdx.x * 64 + (wid & 1) * 32;
  const int z  = blockIdx.z;
  const int zi = z % p.zInner;
  const int zo = z / p.zInner;
  const float* A = p.A + zi * p.sAi + zo * p.sAo;
  const float* B = p.B + zi * p.sBi + zo * p.sBo;
  float*       C = p.C + zi * p.sCi + zo * p.sCo;

  const int half  = lane >> 4;   // 0: K phase {0,1}; 1: K phase {2,3}
  const int idx16 = lane & 15;
  const int kph   = half << 1;

  // A row (address-clamped; no value mask needed)
  const int amc = min(m0 + idx16, p.M - 1);
  // B columns for the two N-subtiles (address-clamped + value mask)
  const int bn0 = n0 + idx16, bn1 = n0 + 16 + idx16;
  const int bc0 = bn0 + p.bOff, bc1 = bn1 + p.bOff;
  const float mB0 = (bc0 >= 0 && bc0 < p.bMax) ? 1.f : 0.f;
  const float mB1 = (bc1 >= 0 && bc1 < p.bMax) ? 1.f : 0.f;
  const int bc0c = min(max(bc0, 0), p.bMax - 1);
  const int bc1c = min(max(bc1, 0), p.bMax - 1);

  const size_t ldb = (size_t)p.ldb;
  const float* ap  = A + (size_t)amc * p.lda + (size_t)kph * KSA;
  const float* bp0 = B + (size_t)kph * ldb + bc0c;
  const float* bp1 = B + (size_t)kph * ldb + bc1c;

  v8f acc0 = {0.f, 0.f, 0.f, 0.f, 0.f, 0.f, 0.f, 0.f};
  v8f acc1 = {0.f, 0.f, 0.f, 0.f, 0.f, 0.f, 0.f, 0.f};

  const int K4 = p.K & ~3;
#pragma unroll 2
  for (int k0 = 0; k0 < K4; k0 += 4) {
    v2f a, b0, b1;
    a.x  = ap[0];       a.y  = ap[KSA];
    b0.x = bp0[0];      b0.y = bp0[ldb];
    b1.x = bp1[0];      b1.y = bp1[ldb];
    b0.x *= mB0; b0.y *= mB0;
    b1.x *= mB1; b1.y *= mB1;
    acc0 = __builtin_amdgcn_wmma_f32_16x16x4_f32(false, a, false, b0, (short)0,
                                                 acc0, false, false);
    acc1 = __builtin_amdgcn_wmma_f32_16x16x4_f32(false, a, false, b1, (short)0,
                                                 acc1, false, false);
    ap  += 4 * KSA;
    bp0 += 4 * ldb;
    bp1 += 4 * ldb;
  }
  if (K4 < p.K) {  // K tail (remainder 1..3), uniform branch; masked loads
    const int ka = K4 + kph;
    v2f a, b0, b1;
    a.x = 0.f; a.y = 0.f; b0.x = 0.f; b0.y = 0.f; b1.x = 0.f; b1.y = 0.f;
    if (ka < p.K) {
      a.x  = ap[0];
      b0.x = bp0[0] * mB0;
      b1.x = bp1[0] * mB1;
    }
    if (ka + 1 < p.K) {
      a.y  = ap[KSA];
      b0.y = bp0[ldb] * mB0;
      b1.y = bp1[ldb] * mB1;
    }
    acc0 = __builtin_amdgcn_wmma_f32_16x16x4_f32(false, a, false, b0, (short)0,
                                                 acc0, false, false);
    acc1 = __builtin_amdgcn_wmma_f32_16x16x4_f32(false, a, false, b1, (short)0,
                                                 acc1, false, false);
  }

  // C/D layout: n = subtile_n0 + (lane&15); VGPR r -> M = m0 + r + 8*(lane>=16)
  const int mb = m0 + half * 8;
#pragma unroll
  for (int s = 0; s < 2; ++s) {
    const v8f acc = s ? acc1 : acc0;
    const int n = n0 + s * 16 + idx16;
    if (n < p.N) {
#pragma unroll
      for (int r = 0; r < 8; ++r) {
        const int m = mb + r;
        if (m < p.M) {
          const size_t o = (size_t)m * p.ldc + n;
          float v = p.alpha * acc[r];
          if (p.beta != 0.f) v += p.beta * C[o];
          if (p.bias)  v += p.bias[m];
          if (p.bias2) v += p.b2s * p.bias2[m];
          C[o] = v;
        }
      }
    }
  }
}

// ===================== elementwise kernels =====================
// Build Hann window, rFFT bases (F x 512) and windowed irFFT bases (512 x F).
__global__ void k_consts(float* win, float* dftr, float* dfti, float* idr, float* idi) {
  const int i = blockIdx.x * blockDim.x + threadIdx.x;
  if (i >= F_ * NF_) return;
  const int f = i / NF_;
  const int n = i - f * NF_;
  const float w = 0.5f - 0.5f * cosf(6.28318530717958647692f * (float)n / (float)NF_);
  if (f == 0) win[n] = w;
  const int   m   = (f * n) & (NF_ - 1);          // exact angle reduction
  const float ang = (float)m * (6.28318530717958647692f / (float)NF_);
  const float c = cosf(ang), s = sinf(ang);
  dftr[i] =  c;                                    // [f*512 + n]
  dfti[i] = -s;
  const float sc = ((f == 0) || (f == NF_ / 2)) ? 1.f : 2.f;
  idr[n * F_ + f] =  w * sc * c * (1.f / (float)NF_);
  idi[n * F_ + f] = -w * sc * s * (1.f / (float)NF_);
}

// frames[bc, n, t] = reflect-padded mix * win[n], laid out [n major, t contiguous]
__global__ void k_frames(const float* __restrict__ mix, const float* __restrict__ win,
                         float* __restrict__ frm) {
  const int i = blockIdx.x * blockDim.x + threadIdx.x;
  if (i >= B_ * C_ * NF_ * T_) return;
  const int t  = i % T_;
  const int n  = (i / T_) % NF_;
  const int bc = i / (T_ * NF_);
  int src = t * HOP_ + n - NF_ / 2;
  if (src < 0)      src = -src;
  if (src >= LEN_)  src = 2 * LEN_ - 2 - src;
  frm[i] = mix[(size_t)bc * LEN_ + src] * win[n];
}

// cLog on a complex tensor (B,2,S): S = channels*F*T
__global__ void k_clog(const float* __restrict__ in, float* __restrict__ out,
                       int S, int total) {
  const int i = blockIdx.x * blockDim.x + threadIdx.x;
  if (i >= total) return;
  const int b = i / S, j = i - b * S;
  const size_t ro = (size_t)(2 * b) * S + j, io = ro + S;
  const float re = d_snap(in[ro]);
  const float im = d_snap(in[io]);
  out[ro] = logf(sqrtf(im * im + re * re + EPS_ * EPS_) + 1.f);
  out[io] = atanf(im / re);
}

// leaky-relu on the REAL half only of a (B,2,S) tensor
__global__ void k_lrelu_real(float* __restrict__ x, int S, int total) {
  const int i = blockIdx.x * blockDim.x + threadIdx.x;
  if (i >= total) return;
  const int b = i / S, j = i - b * S;
  const size_t ro = (size_t)(2 * b) * S + j;
  x[ro] = d_lrelu(x[ro]);
}

// cExp (modExp-based) with optional fused trelu; tensor (B,2,H,F,T) in place
__global__ void k_cexp_trelu(float* __restrict__ x, const float* __restrict__ at,
                             const float* __restrict__ ab, int useTr, int total) {
  const int i = blockIdx.x * blockDim.x + threadIdx.x;
  if (i >= total) return;
  const int S = H_ * FT_;
  const int b = i / S, j = i - b * S;
  const size_t ro = (size_t)(2 * b) * S + j, io = ro + S;
  const float e   = d_modexp(x[ro]) - 1.f;
  const float ang = x[io] - 1.f;
  float cr = e * cosf(ang);
  float ci = e * sinf(ang);
  if (useTr) {
    const int h = j / FT_;
    const int f = (j - h * FT_) / T_;
    const float t00 = at[(0 * H_ + h) * F_ + f];
    const float t01 = at[(1 * H_ + h) * F_ + f];
    const float t10 = at[(2 * H_ + h) * F_ + f];
    const float t11 = at[(3 * H_ + h) * F_ + f];
    const float b0  = ab[(0 * H_ + h) * F_ + f];
    const float b1  = ab[(1 * H_ + h) * F_ + f];
    const float nr = cr * t00 + ci * t01 + b0;
    const float ni = cr * t10 + ci * t11 + b1;
    cr = d_lrelu(nr); ci = d_lrelu(ni);
  }
  x[ro] = cr; x[io] = ci;
}

// out = trelu(cMul(a, b), at, ab); tensors (B,2,H,F,T)
__global__ void k_cmul_trelu(const float* __restrict__ a, const float* __restrict__ bb,
                             float* __restrict__ o, const float* __restrict__ at,
                             const float* __restrict__ ab, int total) {
  const int i = blockIdx.x * blockDim.x + threadIdx.x;
  if (i >= total) return;
  const int S = H_ * FT_;
  const int b = i / S, j = i - b * S;
  const size_t ro = (size_t)(2 * b) * S + j, io = ro + S;
  const float ar = a[ro], ai = a[io];
  const float br = bb[ro], bi = bb[io];
  const float cr = ar * br - ai * bi;
  const float ci = ar * bi + ai * br;
  const int h = j / FT_;
  const int f = (j - h * FT_) / T_;
  const float t00 = at[(0 * H_ + h) * F_ + f];
  const float t01 = at[(1 * H_ + h) * F_ + f];
  const float t10 = at[(2 * H_ + h) * F_ + f];
  const float t11 = at[(3 * H_ + h) * F_ + f];
  const float b0  = ab[(0 * H_ + h) * F_ + f];
  const float b1  = ab[(1 * H_ + h) * F_ + f];
  o[ro] = d_lrelu(cr * t00 + ci * t01 + b0);
  o[io] = d_lrelu(cr * t10 + ci * t11 + b1);
}

// td1[b,p,f,t] = mean_c (ts - tt)
__global__ void k_td1(const float* __restrict__ ts, const float* __restrict__ tt,
                      float* __restrict__ td1, int total) {
  const int i = blockIdx.x * blockDim.x + threadIdx.x;
  if (i >= total) return;
  const int z = i / FT_;     // b*2 + p
  const int j = i - z * FT_;
  float s = 0.f;
#pragma unroll
  for (int c = 0; c < C_; ++c) {
    const size_t o = (size_t)(z * C_ + c) * FT_ + j;
    s += ts[o] - tt[o];
  }
  td1[i] = s * (1.f / (float)C_);
}

// tgate on complex (B,2,F,T) -> (B,F,T)
__global__ void k_tgate(const float* __restrict__ fcx, const float* __restrict__ fgt,
                        const float* __restrict__ fgb, float* __restrict__ fb, int total) {
  const int i = blockIdx.x * blockDim.x + threadIdx.x;
  if (i >= total) return;
  const int b = i / FT_, j = i - b * FT_;
  const int f = j / T_;
  const float fr = fcx[(size_t)(2 * b) * FT_ + j];
  const float fi = fcx[(size_t)(2 * b + 1) * FT_ + j];
  const float gr = d_sigmoid(fr * fgt[0 * F_ + f] + fi * fgt[1 * F_ + f] + fgb[f]);
  const float gi = d_sigmoid(fr * fgt[2 * F_ + f] + fi * fgt[3 * F_ + f] + fgb[F_ + f]);
  fb[i] = gr * gi + (1.f - gr) * (1.f - gi);
}

// td = td1*f + td2*(1-f), f broadcast over re/im
__global__ void k_blend(const float* __restrict__ td1, const float* __restrict__ td2,
                        const float* __restrict__ fb, float* __restrict__ td, int total) {
  const int i = blockIdx.x * blockDim.x + threadIdx.x;
  if (i >= total) return;
  const int z = i / FT_;             // b*2 + p
  const int j = i - z * FT_;
  const float fv = fb[(size_t)(z >> 1) * FT_ + j];
  td[i] = td1[i] * fv + td2[i] * (1.f - fv);
}

// overlap-add + window-sum normalize + crop; gather form (<=2 frames per sample)
__global__ void k_ola(const float* __restrict__ ofr, const float* __restrict__ win,
                      float* __restrict__ out, int total) {
  const int i = blockIdx.x * blockDim.x + threadIdx.x;
  if (i >= total) return;
  const int b = i / LEN_, s = i - b * LEN_;
  const int pos = s + NF_ / 2;
  float acc = 0.f, den = 0.f;
  const int tA = pos / HOP_;
#pragma unroll
  for (int d = 0; d < 2; ++d) {
    const int t = tA - d;
    if (t >= 0 && t < T_) {
      const int n = pos - t * HOP_;
      if (n < NF_) {
        acc += ofr[((size_t)b * NF_ + n) * T_ + t];
        const float w = win[n];
        den += w * w;
      }
    }
  }
  out[i] = acc / fmaxf(den, 1e-11f);
}

// ===================== host-side GEMM plumbing =====================
static void run_gemm(hipStream_t st, const GP& p, int nz) {
  dim3 grid((unsigned)cdiv(p.N, 64), (unsigned)cdiv(p.M, 32), (unsigned)nz);
  if (p.ksA == 8) k_gemm_wmma<8><<<grid, dim3(128, 1, 1), 0, st>>>(p);
  else            k_gemm_wmma<1><<<grid, dim3(128, 1, 1), 0, st>>>(p);
}

static GP base_gp() {
  GP p{};              // value-init: all members zero / null
  p.alpha = 1.f;
  p.zInner = 1;
  p.ksA = 1;
  return p;
}

// real 1x1 channel conv applied to both complex parts: z over (b,p)
static void conv1x1_real(hipStream_t st, const float* W, const float* bias,
                         const float* in, float* out, int Cin, int Cout) {
  GP p = base_gp();
  p.A = W; p.B = in; p.C = out;
  p.M = Cout; p.N = FT_; p.K = Cin;
  p.lda = Cin; p.ldb = FT_; p.ldc = FT_;
  p.zInner = 2 * B_;
  p.sBi = (long long)Cin * FT_; p.sCi = (long long)Cout * FT_;
  p.bias = bias;
  p.bOff = 0; p.bMax = p.N;
  run_gemm(st, p, 2 * B_);
}

// complex 1x1 channel conv: 4 real GEMMs; biases folded as b0 -/+ b1
static void conv1x1_cplx(hipStream_t st, const float* W, const float* bias,
                         const float* in, float* out, int K, int M) {
  const long long inP = (long long)K * FT_, outP = (long long)M * FT_;
  const long long wI  = (long long)M * K;
  const long long wOf[4] = {0, wI, 0, wI};
  const long long bOf[4] = {0, inP, inP, 0};
  const long long cOf[4] = {0, 0, outP, outP};
  const float     al[4]  = {1.f, -1.f, 1.f, 1.f};
  for (int c = 0; c < 4; ++c) {
    GP p = base_gp();
    p.A = W + wOf[c]; p.B = in + bOf[c]; p.C = out + cOf[c];
    p.M = M; p.N = FT_; p.K = K;
    p.lda = K; p.ldb = FT_; p.ldc = FT_;
    p.zInner = B_;
    p.sBi = 2 * inP; p.sCi = 2 * outP;
    p.alpha = al[c];
    const bool first = (c == 0 || c == 2);
    p.beta = first ? 0.f : 1.f;
    if (first) { p.bias = bias; p.bias2 = bias + M; p.b2s = (c == 0) ? -1.f : 1.f; }
    p.bOff = 0; p.bMax = p.N;
    run_gemm(st, p, B_);
  }
}

// causal temporal conv (kernel 8, pad 7 then crop 7) with real weights, both parts
static void tconv_real(hipStream_t st, const float* W, const float* bias,
                       const float* in, float* out, int Cin, int Cout) {
  for (int k = 0; k < 8; ++k) {
    GP p = base_gp();
    p.A = W + k; p.lda = Cin * 8; p.ksA = 8;
    p.B = in; p.C = out;
    p.M = Cout; p.N = T_; p.K = Cin;
    p.ldb = FT_; p.ldc = FT_;
    p.zInner = F_;
    p.sBi = T_; p.sBo = (long long)Cin * FT_;
    p.sCi = T_; p.sCo = (long long)Cout * FT_;
    p.beta = (k == 0) ? 0.f : 1.f;
    if (k == 0) p.bias = bias;
    p.bOff = k - 7; p.bMax = T_;
    run_gemm(st, p, 2 * B_ * F_);
  }
}

// causal temporal complex conv: 4 part-combos x 8 taps
static void tconv_cplx(hipStream_t st, const float* W, const float* bias,
                       const float* in, float* out, int Cin, int Cout) {
  const long long wI  = (long long)Cout * Cin * 8;
  const long long inP = (long long)Cin * FT_, outP = (long long)Cout * FT_;
  const long long wOf[4] = {0, wI, 0, wI};
  const long long bOf[4] = {0, inP, inP, 0};
  const long long cOf[4] = {0, 0, outP, outP};
  const float     al[4]  = {1.f, -1.f, 1.f, 1.f};
  for (int c = 0; c < 4; ++c) {
    for (int k = 0; k < 8; ++k) {
      GP p = base_gp();
      p.A = W + wOf[c] + k; p.lda = Cin * 8; p.ksA = 8;
      p.B = in + bOf[c]; p.C = out + cOf[c];
      p.M = Cout; p.N = T_; p.K = Cin;
      p.ldb = FT_; p.ldc = FT_;
      p.zInner = F_;
      p.sBi = T_; p.sBo = 2 * inP;
      p.sCi = T_; p.sCo = 2 * outP;
      p.alpha = al[c];
      const bool first = (c == 0 || c == 2) && (k == 0);
      p.beta = first ? 0.f : 1.f;
      if (first) { p.bias = bias; p.bias2 = bias + Cout; p.b2s = (c == 0) ? -1.f : 1.f; }
      p.bOff = k - 7; p.bMax = T_;
      run_gemm(st, p, B_ * F_);
    }
  }
}

// frequency-mixing complex "conv" (fs/fr): F x F GEMM per (b, channel)
static void fmix(hipStream_t st, const float* W, const float* bias,
                 const float* in, float* out, int Cch) {
  const long long partP = (long long)Cch * FT_;
  const long long wI    = (long long)F_ * F_;
  const long long wOf[4] = {0, wI, 0, wI};
  const long long bOf[4] = {0, partP, partP, 0};
  const long long cOf[4] = {0, 0, partP, partP};
  const float     al[4]  = {1.f, -1.f, 1.f, 1.f};
  for (int c = 0; c < 4; ++c) {
    GP p = base_gp();
    p.A = W + wOf[c]; p.B = in + bOf[c]; p.C = out + cOf[c];
    p.M = F_; p.N = T_; p.K = F_;
    p.lda = F_; p.ldb = T_; p.ldc = T_;
    p.zInner = Cch;
    p.sBi = FT_; p.sBo = 2 * partP;
    p.sCi = FT_; p.sCo = 2 * partP;
    p.alpha = al[c];
    const bool first = (c == 0 || c == 2);
    p.beta = first ? 0.f : 1.f;
    if (first) { p.bias = bias; p.bias2 = bias + F_; p.b2s = (c == 0) ? -1.f : 1.f; }
    p.bOff = 0; p.bMax = p.N;
    run_gemm(st, p, B_ * Cch);
  }
}

// ===================== entry point =====================
extern "C" void kernel_launch(void* const* d_in, const int* in_sizes, int n_in,
                              void* d_out, int out_size, void* d_ws, size_t ws_size,
                              hipStream_t stream) {
  (void)in_sizes; (void)n_in; (void)out_size;
  const float* mix   = (const float*)d_in[0];
  const float* fs_w  = (const float*)d_in[1];
  const float* fs_b  = (const float*)d_in[2];
  const float* fr_w  = (const float*)d_in[3];
  const float* fr_b  = (const float*)d_in[4];
  const float* c1r_w = (const float*)d_in[5];
  const float* c1r_b = (const float*)d_in[6];
  const float* c1i_w = (const float*)d_in[7];
  const float* c1i_b = (const float*)d_in[8];
  const float* c1_w  = (const float*)d_in[9];
  const float* c1_b  = (const float*)d_in[10];
  const float* c2r_w = (const float*)d_in[11];
  const float* c2r_b = (const float*)d_in[12];
  const float* c2i_w = (const float*)d_in[13];
  const float* c2i_b = (const float*)d_in[14];
  const float* c2_w  = (const float*)d_in[15];
  const float* c2_b  = (const float*)d_in[16];
  const float* a1_t  = (const float*)d_in[17];
  const float* a1_b  = (const float*)d_in[18];
  const float* a2_t  = (const float*)d_in[19];
  const float* a2_b  = (const float*)d_in[20];
  const float* c3r_w = (const float*)d_in[21];
  const float* c3r_b = (const float*)d_in[22];
  const float* c3i_w = (const float*)d_in[23];
  const float* c3i_b = (const float*)d_in[24];
  const float* c3_w  = (const float*)d_in[25];
  const float* c3_b  = (const float*)d_in[26];
  const float* c4r_w = (const float*)d_in[27];
  const float* c4r_b = (const float*)d_in[28];
  const float* c4i_w = (const float*)d_in[29];
  const float* c4i_b = (const float*)d_in[30];
  const float* c4_w  = (const float*)d_in[31];
  const float* c4_b  = (const float*)d_in[32];
  const float* a3_t  = (const float*)d_in[33];
  const float* a3_b  = (const float*)d_in[34];
  const float* a4_t  = (const float*)d_in[35];
  const float* a4_b  = (const float*)d_in[36];
  const float* f1_w  = (const float*)d_in[37];
  const float* f1_b  = (const float*)d_in[38];
  const float* f2_w  = (const float*)d_in[39];
  const float* f2_b  = (const float*)d_in[40];
  const float* cw_w  = (const float*)d_in[41];
  const float* cw_b  = (const float*)d_in[42];
  const float* fg_t  = (const float*)d_in[43];
  const float* fg_b  = (const float*)d_in[44];
  float* out = (float*)d_out;

  // ---- workspace layout (floats) ----
  float* wsp = (float*)d_ws;
  size_t used = 0;
  auto alloc = [&](size_t n) { float* p = wsp + used; used += n; return p; };
  const size_t specC = (size_t)B_ * 2 * C_ * FT_;
  const size_t specH = (size_t)B_ * 2 * H_ * FT_;
  const size_t spec1 = (size_t)B_ * 2 * FT_;
  float* win  = alloc(NF_);
  float* dftr = alloc((size_t)F_ * NF_);
  float* dfti = alloc((size_t)F_ * NF_);
  float* idr  = alloc((size_t)NF_ * F_);
  float* idi  = alloc((size_t)NF_ * F_);
  float* frm  = alloc((size_t)B_ * C_ * NF_ * T_);
  float* sp0  = alloc(specC);   // raw STFT
  float* ts   = alloc(specC);   // after fs mix
  float* cc   = alloc(specC);   // cLog scratch / f1 output
  float* X    = alloc(specH);   // holds tl1 through fuse
  float* Y    = alloc(specH);
  float* Z    = alloc(specH);
  float* td1  = alloc(spec1);
  float* td2  = alloc(spec1);
  float* fcx  = alloc(spec1);
  float* fb   = alloc((size_t)B_ * FT_);
  float* td   = alloc(spec1);
  float* osp  = alloc(spec1);
  float* ofr  = alloc((size_t)B_ * NF_ * T_);
  if (used * sizeof(float) > ws_size) return;   // workspace too small: bail deterministically

  auto g1 = [](int total) { return dim3((unsigned)cdiv(total, 256)); };

  // ---- constants, framing, rFFT (as GEMM vs dense basis) ----
  k_consts<<<g1(F_ * NF_), 256, 0, stream>>>(win, dftr, dfti, idr, idi);
  k_frames<<<g1(B_ * C_ * NF_ * T_), 256, 0, stream>>>(mix, win, frm);
  for (int part = 0; part < 2; ++part) {
    GP p = base_gp();
    p.A = part ? dfti : dftr; p.lda = NF_;
    p.B = frm; p.ldb = T_;
    p.C = sp0 + (size_t)part * C_ * FT_; p.ldc = T_;
    p.M = F_; p.N = T_; p.K = NF_;
    p.zInner = C_;
    p.sBi = (long long)NF_ * T_; p.sBo = (long long)C_ * NF_ * T_;
    p.sCi = FT_;                 p.sCo = 2LL * C_ * FT_;
    p.bOff = 0; p.bMax = p.N;
    run_gemm(stream, p, B_ * C_);
  }
  fmix(stream, fs_w, fs_b, sp0, ts, C_);                               // ts

  const int totH = B_ * H_ * FT_;
  const int totC = B_ * C_ * FT_;

  // ======== branch 1 ========
  k_clog<<<g1(totC), 256, 0, stream>>>(ts, cc, C_ * FT_, totC);
  conv1x1_real(stream, c1r_w, c1r_b, cc, Y, C_, H_);
  k_lrelu_real<<<g1(totH), 256, 0, stream>>>(Y, H_ * FT_, totH);
  conv1x1_real(stream, c1i_w, c1i_b, Y, Z, H_, H_);
  k_cexp_trelu<<<g1(totH), 256, 0, stream>>>(Z, a1_t, a1_b, 1, totH);
  conv1x1_cplx(stream, c1_w, c1_b, Z, Y, H_, H_);
  k_clog<<<g1(totH), 256, 0, stream>>>(Y, Y, H_ * FT_, totH);
  tconv_real(stream, c2r_w, c2r_b, Y, Z, H_, H_);
  k_lrelu_real<<<g1(totH), 256, 0, stream>>>(Z, H_ * FT_, totH);
  tconv_real(stream, c2i_w, c2i_b, Z, X, H_, H_);
  k_cexp_trelu<<<g1(totH), 256, 0, stream>>>(X, nullptr, nullptr, 0, totH);   // X = tl1
  tconv_cplx(stream, c2_w, c2_b, ts, Y, C_, H_);                              // tc
  k_cmul_trelu<<<g1(totH), 256, 0, stream>>>(X, Y, Z, a2_t, a2_b, totH);
  conv1x1_cplx(stream, f1_w, f1_b, Z, cc, H_, C_);
  k_td1<<<g1(B_ * 2 * FT_), 256, 0, stream>>>(ts, cc, td1, B_ * 2 * FT_);

  // ======== branch 2 ========
  k_clog<<<g1(totC), 256, 0, stream>>>(ts, cc, C_ * FT_, totC);
  conv1x1_real(stream, c3r_w, c3r_b, cc, Y, C_, H_);
  k_lrelu_real<<<g1(totH), 256, 0, stream>>>(Y, H_ * FT_, totH);
  conv1x1_real(stream, c3i_w, c3i_b, Y, Z, H_, H_);
  k_cexp_trelu<<<g1(totH), 256, 0, stream>>>(Z, a3_t, a3_b, 1, totH);
  conv1x1_cplx(stream, c3_w, c3_b, Z, Y, H_, H_);
  k_clog<<<g1(totH), 256, 0, stream>>>(Y, Y, H_ * FT_, totH);
  tconv_real(stream, c4r_w, c4r_b, Y, Z, H_, H_);
  k_lrelu_real<<<g1(totH), 256, 0, stream>>>(Z, H_ * FT_, totH);
  tconv_real(stream, c4i_w, c4i_b, Z, Y, H_, H_);
  k_cexp_trelu<<<g1(totH), 256, 0, stream>>>(Y, nullptr, nullptr, 0, totH);   // Y = tl2
  tconv_cplx(stream, c4_w, c4_b, ts, Z, C_, H_);                              // tc2
  k_cmul_trelu<<<g1(totH), 256, 0, stream>>>(Y, Z, Z, a4_t, a4_b, totH);
  conv1x1_cplx(stream, f2_w, f2_b, Z, td2, H_, 1);

  // ======== fuse: f = cconv(concat([tl1, tl2]), cw_w, cw_b); gate; blend ========
  {
    const long long hP = (long long)H_ * FT_;
    const float* Bp[8] = { X,        Y,        X + hP,   Y + hP,
                           X + hP,   Y + hP,   X,        Y };
    const long long wOf[8] = { 0, 64, 128, 192, 0, 64, 128, 192 };
    const float     al[8]  = { 1.f, 1.f, -1.f, -1.f, 1.f, 1.f, 1.f, 1.f };
    for (int c = 0; c < 8; ++c) {
      GP p = base_gp();
      p.A = cw_w + wOf[c]; p.lda = 2 * H_;
      p.B = Bp[c]; p.ldb = FT_;
      p.C = fcx + ((c >= 4) ? FT_ : 0); p.ldc = FT_;
      p.M = 1; p.N = FT_; p.K = H_;
      p.zInner = B_;
      p.sBi = 2 * hP; p.sCi = 2LL * FT_;
      p.alpha = al[c];
      const bool first = (c == 0 || c == 4);
      p.beta = first ? 0.f : 1.f;
      if (first) { p.bias = cw_b; p.bias2 = cw_b + 1; p.b2s = (c == 0) ? -1.f : 1.f; }
      p.bOff = 0; p.bMax = p.N;
      run_gemm(stream, p, B_);
    }
  }
  k_tgate<<<g1(B_ * FT_), 256, 0, stream>>>(fcx, fg_t, fg_b, fb, B_ * FT_);
  k_blend<<<g1(B_ * 2 * FT_), 256, 0, stream>>>(td1, td2, fb, td, B_ * 2 * FT_);

  // ======== fr mix, irFFT (as GEMM), overlap-add ========
  fmix(stream, fr_w, fr_b, td, osp, 1);
  for (int part = 0; part < 2; ++part) {
    GP p = base_gp();
    p.A = part ? idi : idr; p.lda = F_;
    p.B = osp + (size_t)part * FT_; p.ldb = T_;
    p.C = ofr; p.ldc = T_;
    p.M = NF_; p.N = T_; p.K = F_;
    p.zInner = B_;
    p.sBi = 2LL * FT_; p.sCi = (long long)NF_ * T_;
    p.beta = part ? 1.f : 0.f;
    p.bOff = 0; p.bMax = p.N;
    run_gemm(stream, p, B_);
  }
  k_ola<<<g1(B_ * LEN_), 256, 0, stream>>>(ofr, win, out, B_ * LEN_);
}